// TorchModel_120259084518
// MI455X (gfx1250) — compile-verified
//
#include <hip/hip_runtime.h>
#include <hip/hip_bf16.h>
#include <math.h>

typedef __attribute__((ext_vector_type(2))) float v2f;
typedef __attribute__((ext_vector_type(8))) float v8f;

#define Bb 64
#define Ss 512
#define Hh 768
#define Tt 9
#define LDE 16   // padded emission row (9 -> 16)

__device__ __forceinline__ float lane_read(float v, int srcLane) {
  // wave32 cross-lane gather: dst[lane] = v[srcLane]
  return __int_as_float(__builtin_amdgcn_ds_bpermute(srcLane << 2, __float_as_int(v)));
}

// ---------------------------------------------------------------------------
// Kernel 1: emissions[BS][16] = hidden(BS,768) @ W(768,9) + b, N padded to 16.
// One wave computes one 16-row tile with V_WMMA_F32_16X16X4_F32 (full fp32).
// ---------------------------------------------------------------------------
__global__ __launch_bounds__(256) void emis_gemm(const float* __restrict__ hidden,
                                                 const float* __restrict__ W,
                                                 const float* __restrict__ bvec,
                                                 float* __restrict__ E) {
  __shared__ float Wp[Hh * LDE];            // 48 KB, zero-padded columns 9..15
  const int tid = threadIdx.x;
  for (int i = tid; i < Hh * LDE; i += 256) {
    const int k = i >> 4, n = i & 15;
    Wp[i] = (n < Tt) ? W[k * Tt + n] : 0.0f;
  }
  __syncthreads();

  const int lane = tid & 31;
  const int wid  = tid >> 5;
  const int half = lane >> 4;               // 0: K={0,1}, 1: K={2,3}
  const int lr   = lane & 15;               // A: row M ; B/D: col N
  const int rowBase = (blockIdx.x * 8 + wid) * 16;

  // A fragment source: lane (half,lr) reads hidden[rowBase+lr][k0 + 2*half + {0,1}]
  const float* arow = hidden + (size_t)(rowBase + lr) * Hh + half * 2;

  v8f c = {};
#pragma unroll 4
  for (int k0 = 0; k0 < Hh; k0 += 4) {
    v2f a;
    a.x = arow[k0];
    a.y = arow[k0 + 1];
    v2f bf;  // B 4x16: VGPR0 = K={0|2} row, VGPR1 = K={1|3} row, N = lr
    bf.x = Wp[(k0 + 2 * half) * LDE + lr];
    bf.y = Wp[(k0 + 2 * half + 1) * LDE + lr];
    c = __builtin_amdgcn_wmma_f32_16x16x4_f32(false, a, false, bf, (short)0, c,
                                              false, false);
  }

  const float bias = (lr < Tt) ? bvec[lr] : 0.0f;
  const int rb = rowBase + half * 8;        // D: VGPR v -> row v (lanes 0-15) / v+8
#pragma unroll
  for (int v = 0; v < 8; ++v) {
    E[(size_t)(rb + v) * LDE + lr] = c[v] + bias;
  }
}

// ---------------------------------------------------------------------------
// Kernel 2: CRF forward scan. One 16-lane half-wave per batch; lane t owns
// alpha[t]. logsumexp gather via ds_bpermute. Also accumulates the gold-path
// numerator (mask-correct). One wave (2 batches) per block, 32 blocks.
// ---------------------------------------------------------------------------
__global__ __launch_bounds__(32) void crf_scan(const float* __restrict__ E,
                                               const float* __restrict__ startT,
                                               const float* __restrict__ endT,
                                               const float* __restrict__ trans,
                                               const int* __restrict__ labels,
                                               float* __restrict__ res) {
  const int lane = threadIdx.x & 31;
  const int half = lane >> 4;
  const int t    = lane & 15;
  const int base = lane & 16;               // first lane of this batch's group
  const int b    = blockIdx.x * 2 + half;
  const bool act = (t < Tt);

  const float* Erow = E + (size_t)b * Ss * LDE;
  const int*   lb   = labels + b * Ss;

  float trT[Tt];                            // trans[t'][t] for my column t
#pragma unroll
  for (int tp = 0; tp < Tt; ++tp) trT[tp] = act ? trans[tp * Tt + t] : 0.0f;

  float al = act ? (startT[t] + Erow[t]) : -1e30f;

  int prev = lb[0] > 0 ? lb[0] : 0;
  float num = startT[prev] + Erow[prev];

  for (int s = 1; s < Ss; ++s) {
    const float* er = Erow + s * LDE;
    const int lab = lb[s];
    const float mk = (lab > -1) ? 1.0f : 0.0f;
    const int tg = (lab > 0) ? lab : 0;

    // numerator (uniform within half-group; computed by all lanes)
    num += (trans[prev * Tt + tg] + er[tg]) * mk;
    prev = (mk > 0.0f) ? tg : prev;

    // denominator: alpha_new[t] = lse_{t'}(alpha[t'] + trans[t',t]) + em[t]
    const float em = er[t];
    float v[Tt];
    float m = -1e30f;
#pragma unroll
    for (int tp = 0; tp < Tt; ++tp) {
      const float a = lane_read(al, base + tp);
      v[tp] = a + trT[tp];
      m = fmaxf(m, v[tp]);
    }
    float ss = 0.0f;
#pragma unroll
    for (int tp = 0; tp < Tt; ++tp) ss += expf(v[tp] - m);
    const float na = m + logf(ss) + em;
    al = (act && mk > 0.0f) ? na : al;
  }

  // den = logsumexp_t(alpha[t] + end_trans[t])
  float v[Tt];
  float m = -1e30f;
#pragma unroll
  for (int tp = 0; tp < Tt; ++tp) {
    const float a = lane_read(al, base + tp);
    v[tp] = a + endT[tp];
    m = fmaxf(m, v[tp]);
  }
  float ss = 0.0f;
#pragma unroll
  for (int tp = 0; tp < Tt; ++tp) ss += expf(v[tp] - m);
  const float den = m + logf(ss);
  num += endT[prev];

  if (t == 0) res[b] = den - num;
}

// ---------------------------------------------------------------------------
// Kernel 3: out = mean(res[0..63]) via wave32 butterfly reduction.
// ---------------------------------------------------------------------------
__global__ __launch_bounds__(32) void reduce_mean(const float* __restrict__ res,
                                                  float* __restrict__ out) {
  const int lane = threadIdx.x;
  float v = res[lane] + res[lane + 32];
#pragma unroll
  for (int off = 16; off >= 1; off >>= 1) v += lane_read(v, lane ^ off);
  if (lane == 0) out[0] = v * (1.0f / 64.0f);
}

extern "C" void kernel_launch(void* const* d_in, const int* in_sizes, int n_in,
                              void* d_out, int out_size, void* d_ws, size_t ws_size,
                              hipStream_t stream) {
  const float* hidden = (const float*)d_in[0];
  const float* W      = (const float*)d_in[1];
  const float* bvec   = (const float*)d_in[2];
  const float* startT = (const float*)d_in[3];
  const float* endT   = (const float*)d_in[4];
  const float* trans  = (const float*)d_in[5];
  const int*   labels = (const int*)d_in[6];
  float* out = (float*)d_out;

  float* E   = (float*)d_ws;                        // [64*512][16] f32 = 2 MB
  float* res = E + (size_t)Bb * Ss * LDE;           // [64] f32

  emis_gemm<<<(Bb * Ss) / 128, 256, 0, stream>>>(hidden, W, bvec, E);
  crf_scan<<<Bb / 2, 32, 0, stream>>>(E, startT, endT, trans, labels, res);
  reduce_mean<<<1, 32, 0, stream>>>(res, out);
}